// BGEM3Model_43284680409450
// MI455X (gfx1250) — compile-verified
//
#include <hip/hip_runtime.h>

typedef __attribute__((ext_vector_type(2))) float v2f;
typedef __attribute__((ext_vector_type(8))) float v8f;

#define B_DIM 32
#define S_DIM 1024
#define H_DIM 1024
#define V_DIM 250002

// ---------------------------------------------------------------------------
// Kernel 1: zero-fill the [B, V] output table (harness poisons it with 0xAA).
// ---------------------------------------------------------------------------
__global__ void zero_out_kernel(float4* __restrict__ out4, int n4,
                                float* __restrict__ out, int n) {
  int i = blockIdx.x * blockDim.x + threadIdx.x;
  const int stride = gridDim.x * blockDim.x;
  const float4 z = make_float4(0.f, 0.f, 0.f, 0.f);
  for (int j = i; j < n4; j += stride) out4[j] = z;
  // scalar tail (n is divisible by 4 here, but stay general)
  for (int j = n4 * 4 + i; j < n; j += stride) out[j] = 0.f;
}

// ---------------------------------------------------------------------------
// Kernel 2: fused GEMV (via V_WMMA_F32_16X16X4_F32) + ReLU + scatter-max.
// One wave owns 16 rows of hidden_state. Each WMMA consumes a 16x4 f32 tile
// of A (hidden rows) against B = w[k..k+3] broadcast across all 16 columns,
// accumulating the dot products in the f32 C/D tile.
//
// A layout (ISA 7.12.2, 32-bit 16x4): lane<16  -> (A[m][k],   A[m][k+1])
//                                     lane>=16 -> (A[m][k+2], A[m][k+3])
// i.e. a contiguous float2 at column offset koff = 2*(lane>=16).
// B 4x16 uses the same half-split K striping -> same float2 of w at koff.
// C/D: lane 0 holds rows 0..7 (v[0..7]), lane 16 holds rows 8..15.
// ---------------------------------------------------------------------------
__global__ __launch_bounds__(256)
void sparse_embed_kernel(const float* __restrict__ hs,   // [B*S, H]
                         const int*   __restrict__ ids,  // [B*S]
                         const float* __restrict__ w,    // [H]
                         const float* __restrict__ bias, // [1]
                         float*       __restrict__ out)  // [B, V]
{
  __shared__ float wl[H_DIM];

  const int tid = threadIdx.x;
  // Stage w into LDS: 256 threads x float4 = 1024 floats.
  ((float4*)wl)[tid] = ((const float4*)w)[tid];
  __syncthreads();

  const int lane = tid & 31;
  const int wave = tid >> 5;                       // 8 waves / block
  const int row_base = (blockIdx.x * 8 + wave) * 16;
  const int m    = lane & 15;
  const int koff = (lane >> 4) << 1;               // 0 for lanes 0-15, 2 for 16-31

  const float* rowp = hs + (size_t)(row_base + m) * H_DIM + koff;
  const float* wp   = wl + koff;

  v8f c0 = {};
  v8f c1 = {};
  #pragma unroll 4
  for (int k = 0; k < H_DIM; k += 8) {
    v2f a0 = *(const v2f*)(rowp + k);
    v2f b0 = *(const v2f*)(wp + k);
    c0 = __builtin_amdgcn_wmma_f32_16x16x4_f32(false, a0, false, b0,
                                               (short)0, c0, false, false);
    v2f a1 = *(const v2f*)(rowp + k + 4);
    v2f b1 = *(const v2f*)(wp + k + 4);
    c1 = __builtin_amdgcn_wmma_f32_16x16x4_f32(false, a1, false, b1,
                                               (short)0, c1, false, false);
  }
  v8f c = c0 + c1;

  const float bv = bias[0];

  // Lane 0 scatters rows row_base..+7, lane 16 scatters rows row_base+8..+15.
  if ((lane & 15) == 0) {
    const int rbase = row_base + (lane >> 4) * 8;
    #pragma unroll
    for (int j = 0; j < 8; ++j) {
      float tw = c[j] + bv;
      if (tw > 0.0f) {                 // ReLU; zero scatters are no-ops anyway
        const int r  = rbase + j;
        const int id = ids[r];
        if (id > 3) {                  // columns 0..3 forced to zero by reference
          const int b = r >> 10;       // r / S_DIM
          // values are non-negative -> u32 bit pattern is order-preserving
          atomicMax((unsigned int*)(out + (size_t)b * V_DIM + id),
                    __float_as_uint(tw));
        }
      }
    }
  }
}

// ---------------------------------------------------------------------------
// Launch: inputs are (hidden_state f32 [B*S*H], input_ids i32 [B*S],
//                     W_sparse f32 [H], b_sparse f32 [1]); output f32 [B*V].
// ---------------------------------------------------------------------------
extern "C" void kernel_launch(void* const* d_in, const int* in_sizes, int n_in,
                              void* d_out, int out_size, void* d_ws, size_t ws_size,
                              hipStream_t stream) {
  const float* hs   = (const float*)d_in[0];
  const int*   ids  = (const int*)  d_in[1];
  const float* w    = (const float*)d_in[2];
  const float* bias = (const float*)d_in[3];
  float* out = (float*)d_out;

  const int n  = out_size;          // 8,000,064
  const int n4 = n / 4;             // 2,000,016
  zero_out_kernel<<<2048, 256, 0, stream>>>((float4*)out, n4, out, n);

  // 32768 rows / (8 waves * 16 rows) = 256 blocks
  sparse_embed_kernel<<<256, 256, 0, stream>>>(hs, ids, w, bias, out);
}